// DeformableTransformerEncoderLayer_31851477467859
// MI455X (gfx1250) — compile-verified
//
#include <hip/hip_runtime.h>
#include <math.h>

typedef __attribute__((ext_vector_type(2))) float v2f;
typedef __attribute__((ext_vector_type(8))) float v8f;
typedef int v4i __attribute__((vector_size(16)));   // matches builtin param type

#define DM   256
#define NH   8
#define HD   32
#define NL   4
#define NP   4
#define BQ   4
#define LQN  5440
#define MTOT (BQ * LQN)   // 21760 rows total

#define GEMM_WAVES 4      // waves per block
#define MREP 4            // 16-row m-tiles per wave
#define MB_BLOCKS (MTOT / (16 * GEMM_WAVES * MREP))  // 85

#if __has_builtin(__builtin_amdgcn_global_load_async_to_lds_b128) && \
    __has_builtin(__builtin_amdgcn_s_wait_asynccnt)
#define USE_ASYNC_LDS 1
typedef __attribute__((address_space(1))) v4i* gv4i_p;
typedef __attribute__((address_space(3))) v4i* lv4i_p;
#else
#define USE_ASYNC_LDS 0
#endif

__device__ __forceinline__ int imin(int a, int b) { return a < b ? a : b; }
__device__ __forceinline__ int imax(int a, int b) { return a > b ? a : b; }

// ---------------------------------------------------------------------------
// q = src + pos   (vectorized float4)
// ---------------------------------------------------------------------------
__global__ void add_pos_kernel(const float* __restrict__ src,
                               const float* __restrict__ pos,
                               float* __restrict__ q, int n4) {
  int i = blockIdx.x * blockDim.x + threadIdx.x;
  if (i < n4) {
    float4 a = ((const float4*)src)[i];
    float4 b = ((const float4*)pos)[i];
    ((float4*)q)[i] = make_float4(a.x + b.x, a.y + b.y, a.z + b.z, a.w + b.w);
  }
}

// ---------------------------------------------------------------------------
// fp32 WMMA GEMM: C[M,N] = A[M,256] @ W[256,N] + bias (+R) (+relu), N known
// at compile time so all strided offsets fold into immediate offsets.
// Block: 4 waves sharing one 16-col n-tile (W tile staged in LDS via
// GLOBAL_LOAD_ASYNC_TO_LDS_B128 on CDNA5), each wave computing a 64x16
// output (4 WMMA per K4-step).
// ---------------------------------------------------------------------------
template <int N, int TILESN>
__launch_bounds__(GEMM_WAVES * 32)
__global__ void wmma_gemm_kernel(const float* __restrict__ A,
                                 const float* __restrict__ W,
                                 const float* __restrict__ bias,
                                 const float* __restrict__ R,  // residual/null
                                 float* __restrict__ C,
                                 int do_relu) {
  __shared__ float ldsW[DM * 16];  // W[:, n0:n0+16] row-major, 16 KB

  const int tid  = threadIdx.x;
  const int wave = tid >> 5;
  const int lane = tid & 31;
  const int half = lane >> 4;   // 0: K=0,1 ; 1: K=2,3 within each K4 step
  const int l16  = lane & 15;

  const int nb = blockIdx.x % TILESN;
  const int mb = blockIdx.x / TILESN;
  const int n0 = nb * 16;
  const int mtile0 = mb * (GEMM_WAVES * MREP) + wave * MREP;

  // ---- stage W[:, n0:n0+16] into LDS (row-major [k][16]) ----
  {
#if USE_ASYNC_LDS
    // 256 rows x 64B; each thread issues 8 async 16B copies.
#pragma unroll
    for (int it = 0; it < 8; ++it) {
      int f   = tid + it * (GEMM_WAVES * 32);  // float4 index, 0..1023
      int row = f >> 2;
      int c4  = f & 3;
      __builtin_amdgcn_global_load_async_to_lds_b128(
          (gv4i_p)(W + (size_t)row * N + n0 + c4 * 4),
          (lv4i_p)(ldsW + row * 16 + c4 * 4), 0, 0);
    }
    __builtin_amdgcn_s_wait_asynccnt(0);
#else
#pragma unroll
    for (int it = 0; it < 8; ++it) {
      int f   = tid + it * (GEMM_WAVES * 32);
      int row = f >> 2;
      int c4  = f & 3;
      float4 v = *(const float4*)(W + (size_t)row * N + n0 + c4 * 4);
      *(float4*)(ldsW + row * 16 + c4 * 4) = v;
    }
#endif
    __syncthreads();
  }

  // A 16x4 fp32 layout: v0 = A[m][k+2*half], v1 = A[m][k+2*half+1]
  const float* Abase = A + (size_t)(mtile0 * 16 + l16) * DM + 2 * half;
  // B 4x16 fp32 layout: v0 = W[k+2*half][n], v1 = W[k+2*half+1][n]
  const float* ldsB = ldsW + half * 32 + l16;

  v8f acc[MREP];
#pragma unroll
  for (int mi = 0; mi < MREP; ++mi)
    acc[mi] = (v8f){0.f, 0.f, 0.f, 0.f, 0.f, 0.f, 0.f, 0.f};

#pragma unroll
  for (int k = 0; k < DM; k += 4) {
    v2f bvv;
    bvv.x = ldsB[k * 16];
    bvv.y = ldsB[k * 16 + 16];
#pragma unroll
    for (int mi = 0; mi < MREP; ++mi) {
      float2 a2 = *(const float2*)(Abase + mi * 16 * DM + k);
      v2f av; av.x = a2.x; av.y = a2.y;
      acc[mi] = __builtin_amdgcn_wmma_f32_16x16x4_f32(
          false, av, false, bvv, (short)0, acc[mi], false, false);
    }
  }

  // C/D layout: acc[mi][i] -> row mtile0*16 + mi*16 + i + 8*half, col n0+l16
  float bcol = bias[n0 + l16];
#pragma unroll
  for (int mi = 0; mi < MREP; ++mi) {
#pragma unroll
    for (int i = 0; i < 8; ++i) {
      int row = (mtile0 + mi) * 16 + i + 8 * half;
      size_t off = (size_t)row * N + n0 + l16;
      float v = acc[mi][i] + bcol;
      if (R) v += R[off];
      if (do_relu) v = fmaxf(v, 0.f);
      C[off] = v;
    }
  }
}

// ---------------------------------------------------------------------------
// Softmax over the last 16 (NL*NP) logits, one thread per (b,l,h) row.
// ---------------------------------------------------------------------------
__global__ void softmax16_kernel(float* __restrict__ aw, int rows) {
  int r = blockIdx.x * blockDim.x + threadIdx.x;
  if (r >= rows) return;
  float* p = aw + (size_t)r * 16;
  float v[16];
  float m = -1e30f;
#pragma unroll
  for (int i = 0; i < 16; ++i) { v[i] = p[i]; m = fmaxf(m, v[i]); }
  float s = 0.f;
#pragma unroll
  for (int i = 0; i < 16; ++i) { v[i] = __expf(v[i] - m); s += v[i]; }
  float inv = 1.f / s;
#pragma unroll
  for (int i = 0; i < 16; ++i) p[i] = v[i] * inv;
}

// ---------------------------------------------------------------------------
// Deformable attention sampling: one wave32 per (b,l,h), lane = head channel.
// ---------------------------------------------------------------------------
__launch_bounds__(256)
__global__ void sampling_kernel(const float* __restrict__ V,
                                const float* __restrict__ OFFp,
                                const float* __restrict__ AW,
                                const float* __restrict__ ref,
                                float* __restrict__ ATT) {
  int wave = threadIdx.x >> 5;
  int d    = threadIdx.x & 31;
  int w    = blockIdx.x * 8 + wave;   // (b*LQ + l)*NH + h
  int h    = w & (NH - 1);
  int ml   = w >> 3;                  // b*LQ + l
  int b    = ml / LQN;

  const float* offp = OFFp + (size_t)ml * DM + h * (NL * NP * 2);
  const float* awp  = AW   + (size_t)ml * (NH * NL * NP) + h * (NL * NP);
  const float* refp = ref  + (size_t)ml * NL * 2;
  const float* Vb   = V + (size_t)b * LQN * DM + h * HD + d;

  const int dims[4]  = {64, 32, 16, 8};
  const int start[4] = {0, 4096, 5120, 5376};

  float acc = 0.f;
#pragma unroll
  for (int lev = 0; lev < NL; ++lev) {
    const int Hl = dims[lev], Wl = dims[lev], s = start[lev];
    const float invW = 1.f / (float)Wl, invH = 1.f / (float)Hl;
    float rx = refp[lev * 2 + 0], ry = refp[lev * 2 + 1];
#pragma unroll
    for (int p = 0; p < NP; ++p) {
      float ox = offp[(lev * NP + p) * 2 + 0];
      float oy = offp[(lev * NP + p) * 2 + 1];
      float a  = awp[lev * NP + p];
      float x = (rx + ox * invW) * (float)Wl - 0.5f;
      float y = (ry + oy * invH) * (float)Hl - 0.5f;
      float x0f = floorf(x), y0f = floorf(y);
      float fx = x - x0f, fy = y - y0f;
      int x0 = (int)x0f, y0 = (int)y0f;
#pragma unroll
      for (int dy = 0; dy < 2; ++dy) {
#pragma unroll
        for (int dx = 0; dx < 2; ++dx) {
          int xi = x0 + dx, yi = y0 + dy;
          float wgt = (dx ? fx : 1.f - fx) * (dy ? fy : 1.f - fy);
          bool valid = (xi >= 0) && (xi < Wl) && (yi >= 0) && (yi < Hl);
          int xc = imin(imax(xi, 0), Wl - 1);
          int yc = imin(imax(yi, 0), Hl - 1);
          float g = Vb[(size_t)(s + yc * Wl + xc) * DM];
          acc += valid ? g * wgt * a : 0.f;
        }
      }
    }
  }
  ATT[(size_t)ml * DM + h * HD + d] = acc;
}

// ---------------------------------------------------------------------------
// LayerNorm over D=256: one wave32 per row, 8 elems/lane, shfl_xor reductions.
// ---------------------------------------------------------------------------
__launch_bounds__(256)
__global__ void layernorm_kernel(const float* __restrict__ X,
                                 const float* __restrict__ g,
                                 const float* __restrict__ bta,
                                 float* __restrict__ Y) {
  int wave = threadIdx.x >> 5;
  int lane = threadIdx.x & 31;
  size_t row = (size_t)blockIdx.x * 8 + wave;
  const float* x = X + row * DM + lane * 8;
  float v[8];
  float s = 0.f;
#pragma unroll
  for (int i = 0; i < 8; ++i) { v[i] = x[i]; s += v[i]; }
#pragma unroll
  for (int m = 16; m >= 1; m >>= 1) s += __shfl_xor(s, m, 32);
  float mean = s * (1.f / 256.f);
  float vs = 0.f;
#pragma unroll
  for (int i = 0; i < 8; ++i) { float dlt = v[i] - mean; vs += dlt * dlt; }
#pragma unroll
  for (int m = 16; m >= 1; m >>= 1) vs += __shfl_xor(vs, m, 32);
  float inv = rsqrtf(vs * (1.f / 256.f) + 1e-5f);
  float* y = Y + row * DM + lane * 8;
#pragma unroll
  for (int i = 0; i < 8; ++i) {
    int c = lane * 8 + i;
    y[i] = (v[i] - mean) * inv * g[c] + bta[c];
  }
}

// ---------------------------------------------------------------------------
extern "C" void kernel_launch(void* const* d_in, const int* in_sizes, int n_in,
                              void* d_out, int out_size, void* d_ws, size_t ws_size,
                              hipStream_t stream) {
  (void)in_sizes; (void)n_in; (void)out_size; (void)ws_size;

  const float* src   = (const float*)d_in[0];
  const float* pos   = (const float*)d_in[1];
  const float* ref   = (const float*)d_in[2];
  const float* Wv    = (const float*)d_in[5];
  const float* bv    = (const float*)d_in[6];
  const float* Woff  = (const float*)d_in[7];
  const float* boff  = (const float*)d_in[8];
  const float* Wattn = (const float*)d_in[9];
  const float* battn = (const float*)d_in[10];
  const float* Wout  = (const float*)d_in[11];
  const float* bout  = (const float*)d_in[12];
  const float* g1    = (const float*)d_in[13];
  const float* b1    = (const float*)d_in[14];
  const float* W1    = (const float*)d_in[15];
  const float* bb1   = (const float*)d_in[16];
  const float* W2    = (const float*)d_in[17];
  const float* bb2   = (const float*)d_in[18];
  const float* g3    = (const float*)d_in[19];
  const float* b3    = (const float*)d_in[20];
  float* out = (float*)d_out;

  const size_t MD = (size_t)MTOT * DM;   // 21760*256 floats
  float* Q    = (float*)d_ws;            // q = src+pos
  float* V    = Q + MD;                  // value
  float* OFFb = V + MD;                  // sampling offsets
  float* AW   = OFFb + MD;               // attention weights (MTOT*128)
  // dead-buffer aliases (sequentially safe)
  float* ATT = Q;      // attended  (q dead after off/attn GEMMs)
  float* X1  = V;      // src + src2 (value dead after sampling)
  float* SXY = OFFb;   // LN1 output (off dead after sampling)
  float* H   = Q;      // FFN hidden (attended dead after Wout GEMM)
  float* X2  = V;      // sxy + ffn  (X1 dead after LN1)

  const int g256 = MB_BLOCKS * 16;  // 85 * 16 = 1360 blocks, N=256
  const int g128 = MB_BLOCKS * 8;   // 85 * 8  = 680  blocks, N=128
  const int gthreads = GEMM_WAVES * 32;

  // 1. q = src + pos
  int n4 = (int)(MD / 4);
  add_pos_kernel<<<(n4 + 255) / 256, 256, 0, stream>>>(src, pos, Q, n4);
  // 2. value = src @ Wv + bv
  wmma_gemm_kernel<256, 16><<<g256, gthreads, 0, stream>>>(src, Wv, bv, nullptr, V, 0);
  // 3. off = q @ Woff + boff
  wmma_gemm_kernel<256, 16><<<g256, gthreads, 0, stream>>>(Q, Woff, boff, nullptr, OFFb, 0);
  // 4. attn logits = q @ Wattn + battn
  wmma_gemm_kernel<128, 8><<<g128, gthreads, 0, stream>>>(Q, Wattn, battn, nullptr, AW, 0);
  // 5. softmax over 16 (per b,l,h)
  int rows = MTOT * NH;
  softmax16_kernel<<<(rows + 255) / 256, 256, 0, stream>>>(AW, rows);
  // 6. attended = deformable sampling
  sampling_kernel<<<MTOT, 256, 0, stream>>>(V, OFFb, AW, ref, ATT);
  // 7. X1 = attended @ Wout + bout + src (residual)
  wmma_gemm_kernel<256, 16><<<g256, gthreads, 0, stream>>>(ATT, Wout, bout, src, X1, 0);
  // 8. sxy = LN(X1; g1, b1)
  layernorm_kernel<<<MTOT / 8, 256, 0, stream>>>(X1, g1, b1, SXY);
  // 9. h = relu(sxy @ W1 + bb1)
  wmma_gemm_kernel<256, 16><<<g256, gthreads, 0, stream>>>(SXY, W1, bb1, nullptr, H, 1);
  // 10. X2 = h @ W2 + bb2 + sxy (residual)
  wmma_gemm_kernel<256, 16><<<g256, gthreads, 0, stream>>>(H, W2, bb2, SXY, X2, 0);
  // 11. out = LN(X2; g3, b3)
  layernorm_kernel<<<MTOT / 8, 256, 0, stream>>>(X2, g3, b3, out);
}